// MRC_20091857011157
// MI455X (gfx1250) — compile-verified
//
#include <hip/hip_runtime.h>

typedef __attribute__((ext_vector_type(2))) float v2f;
typedef __attribute__((ext_vector_type(8))) float v8f;

#define Bx 8
#define Nx 512
#define Dx 768
#define Lx 16
#define SPAN_ELEMS 33554432u   // B*L*N*N
#define SPAN_GROUPS 8388608u   // SPAN_ELEMS/4
#define SPAN_BLOCKS 4096

// ---------------------------------------------------------------------------
// Kernel 1: fused GEMM  [4096 x 768] x [768 x 64]  via V_WMMA_F32_16X16X4_F32.
// Columns 0..31  = x @ W_se   (+ b_se)           -> ws_se  [4096][32] (logits)
// Columns 32..47 = x @ W_span[:D]  (+ b_span)    -> aT     [B][L][N]
// Columns 48..63 = x @ W_span[D:]                -> bbT    [B][L][N]
// One wave per 16x16 output tile; 4 waves/block (one per column tile).
// ---------------------------------------------------------------------------
__global__ __launch_bounds__(128) void gemm_wmma_kernel(
    const float* __restrict__ x, const float* __restrict__ W_se,
    const float* __restrict__ b_se, const float* __restrict__ W_span,
    const float* __restrict__ b_span, float* __restrict__ ws_se,
    float* __restrict__ aT, float* __restrict__ bbT)
{
    const int lane  = threadIdx.x & 31;
    const int t     = threadIdx.x >> 5;   // column tile 0..3 (wave-uniform)
    const int half  = lane >> 4;          // K-half select for A/B fragments
    const int mr    = lane & 15;          // row (A) / column (B, D) index
    const int grow0 = blockIdx.x * 16;    // global row base (tile of 16 rows)

    const float* xrow = x + (size_t)(grow0 + mr) * Dx;

    const float* Wb; int wstride; int coff;
    if (t == 0)      { Wb = W_se;             wstride = 32; coff = mr;      }
    else if (t == 1) { Wb = W_se;             wstride = 32; coff = 16 + mr; }
    else if (t == 2) { Wb = W_span;           wstride = 16; coff = mr;      }
    else             { Wb = W_span + Dx * 16; wstride = 16; coff = mr;      }

    v8f acc = {0.f, 0.f, 0.f, 0.f, 0.f, 0.f, 0.f, 0.f};
    for (int k = 0; k < Dx; k += 4) {
        const int kk = k + 2 * half;
        // A fragment: lane holds row mr, K = kk, kk+1
        v2f af; af.x = xrow[kk]; af.y = xrow[kk + 1];
        // B fragment (mirror layout): lane holds column coff, K = kk, kk+1
        v2f bf; bf.x = Wb[(size_t)kk * wstride + coff];
                bf.y = Wb[(size_t)(kk + 1) * wstride + coff];
        acc = __builtin_amdgcn_wmma_f32_16x16x4_f32(
            /*neg_a=*/false, af, /*neg_b=*/false, bf,
            /*c_mod=*/(short)0, acc, /*reuse_a=*/false, /*reuse_b=*/false);
    }

    // D layout: VGPR r -> M = r + 8*half, N = mr
    #pragma unroll
    for (int r = 0; r < 8; ++r) {
        const int m    = r + 8 * half;
        const int grow = grow0 + m;          // global row = b*512 + n
        const int bidx = grow >> 9;
        const int nidx = grow & 511;
        const float v  = acc[r];
        if (t < 2) {
            const int c = t * 16 + mr;
            ws_se[(size_t)grow * 32 + c] = v + b_se[c];
        } else if (t == 2) {
            aT[(size_t)(bidx * Lx + mr) * Nx + nidx] = v + b_span[mr];
        } else {
            bbT[(size_t)(bidx * Lx + mr) * Nx + nidx] = v;
        }
    }
}

// ---------------------------------------------------------------------------
// Kernel 2: start/end focal-loss partial sums over [B, 2L, N] = 131072 elems.
// ---------------------------------------------------------------------------
__global__ __launch_bounds__(256) void se_loss_kernel(
    const float* __restrict__ ws_se, const float* __restrict__ start,
    const float* __restrict__ endp, const int* __restrict__ seqlen,
    float* __restrict__ seLossP)
{
    __shared__ float sf[256];
    const int g  = blockIdx.x * 256 + threadIdx.x;   // 0..32767
    const int f0 = g * 4;
    const int b  = f0 >> 14;          // 2L*N = 16384 per batch
    const int c  = (f0 >> 9) & 31;    // channel 0..31
    const int n0 = f0 & 511;
    const int sl = seqlen[b];
    const float* yrow = (c < 16) ? (start + (size_t)(b * 16 + c) * Nx)
                                 : (endp  + (size_t)(b * 16 + (c - 16)) * Nx);
    float acc = 0.f;
    #pragma unroll
    for (int e = 0; e < 4; ++e) {
        const int n   = n0 + e;
        const float z = ws_se[(size_t)(b * Nx + n) * 32 + c];
        const float p = 1.f / (1.f + __expf(-z));
        const float q = 1.f - p;
        const float y = yrow[n];
        const float fl = (y == 1.f) ? (-0.5f * q * q * __logf(p))
                                    : (-0.5f * p * p * __logf(q));
        acc += (n < sl) ? fl : 0.f;
    }
    sf[threadIdx.x] = acc; __syncthreads();
    for (int off = 128; off > 0; off >>= 1) {
        if (threadIdx.x < off) sf[threadIdx.x] += sf[threadIdx.x + off];
        __syncthreads();
    }
    if (threadIdx.x == 0) seLossP[blockIdx.x] = sf[0];
}

// ---------------------------------------------------------------------------
// Kernel 3: span branch — the memory-bound hot loop over 33.5M elements.
// 16B vector loads/stores, grid-stride, exact integer count accumulation.
// ---------------------------------------------------------------------------
__device__ __forceinline__ void span_elem(
    float aval, float bbv, int s, int v, float& pr, float& loss,
    unsigned& tp, unsigned& tn, unsigned& fp, unsigned& vs)
{
    const float z = aval + bbv;
    const float p = 1.f / (1.f + __expf(-z));
    const bool  P = z > 0.f;             // sigmoid(z) > 0.5  <=>  z > 0
    pr = P ? 1.f : 0.f;
    if (s == 1) { tp += (unsigned)P; tn += (unsigned)!P; }
    else if ((v == 1) & P) fp += 1u;
    const float q  = 1.f - p;
    const float fl = (s == 1) ? (-0.5f * q * q * __logf(p))
                              : (-0.5f * p * p * __logf(q));
    loss += fl * (float)v;
    vs   += (unsigned)v;
}

__global__ __launch_bounds__(256) void span_kernel(
    const int* __restrict__ span, const int* __restrict__ val,
    const float* __restrict__ aT, const float* __restrict__ bbT,
    float* __restrict__ pred, float* __restrict__ lossP,
    unsigned* __restrict__ cntP)
{
    __shared__ float sf[256];
    __shared__ unsigned su[256];
    float loss = 0.f;
    unsigned tp = 0, tn = 0, fp = 0, vs = 0;

    const unsigned tid0 = blockIdx.x * 256u + threadIdx.x;
    for (unsigned g = tid0; g < SPAN_GROUPS; g += SPAN_BLOCKS * 256u) {
        const int      j0  = (int)(g & 127u) << 2;   // j base (4 per group)
        const unsigned row = g >> 7;                 // (b*L + l)*N + i
        const int      i   = (int)(row & 511u);
        const int      bl  = (int)(row >> 9);        // b*L + l
        const float aval = aT[(size_t)bl * Nx + i];  // includes b_span[l]
        const float4 bb4 = *(const float4*)(bbT + (size_t)bl * Nx + j0);
        const int4   sp4 = *(const int4*)(span + (size_t)g * 4);
        const int4   vv4 = *(const int4*)(val  + (size_t)g * 4);
        float4 pr;
        span_elem(aval, bb4.x, sp4.x, vv4.x, pr.x, loss, tp, tn, fp, vs);
        span_elem(aval, bb4.y, sp4.y, vv4.y, pr.y, loss, tp, tn, fp, vs);
        span_elem(aval, bb4.z, sp4.z, vv4.z, pr.z, loss, tp, tn, fp, vs);
        span_elem(aval, bb4.w, sp4.w, vv4.w, pr.w, loss, tp, tn, fp, vs);
        *(float4*)(pred + (size_t)g * 4) = pr;
    }

    // deterministic block reductions
    sf[threadIdx.x] = loss; __syncthreads();
    for (int off = 128; off > 0; off >>= 1) {
        if (threadIdx.x < off) sf[threadIdx.x] += sf[threadIdx.x + off];
        __syncthreads();
    }
    if (threadIdx.x == 0) lossP[blockIdx.x] = sf[0];
    __syncthreads();

    unsigned cnts[4] = {tp, tn, fp, vs};
    #pragma unroll
    for (int k = 0; k < 4; ++k) {
        su[threadIdx.x] = cnts[k]; __syncthreads();
        for (int off = 128; off > 0; off >>= 1) {
            if (threadIdx.x < off) su[threadIdx.x] += su[threadIdx.x + off];
            __syncthreads();
        }
        if (threadIdx.x == 0) cntP[blockIdx.x * 4 + k] = su[0];
        __syncthreads();
    }
}

// ---------------------------------------------------------------------------
// Kernel 4: final single-block reduction + scalar outputs.
// ---------------------------------------------------------------------------
__global__ __launch_bounds__(256) void final_kernel(
    const float* __restrict__ lossP, const unsigned* __restrict__ cntP,
    const float* __restrict__ seLossP, const int* __restrict__ seqlen,
    float* __restrict__ out_scalars)
{
    __shared__ float sf[256];
    __shared__ unsigned su[256];
    float loss = 0.f, sel = 0.f;
    unsigned tp = 0, tn = 0, fp = 0, vs = 0;
    for (int i = threadIdx.x; i < SPAN_BLOCKS; i += 256) {
        loss += lossP[i];
        tp += cntP[i * 4 + 0]; tn += cntP[i * 4 + 1];
        fp += cntP[i * 4 + 2]; vs += cntP[i * 4 + 3];
    }
    if (threadIdx.x < 128) sel = seLossP[threadIdx.x];

    sf[threadIdx.x] = loss; __syncthreads();
    for (int off = 128; off > 0; off >>= 1) {
        if (threadIdx.x < off) sf[threadIdx.x] += sf[threadIdx.x + off];
        __syncthreads();
    }
    loss = sf[0]; __syncthreads();

    sf[threadIdx.x] = sel; __syncthreads();
    for (int off = 128; off > 0; off >>= 1) {
        if (threadIdx.x < off) sf[threadIdx.x] += sf[threadIdx.x + off];
        __syncthreads();
    }
    sel = sf[0]; __syncthreads();

    unsigned cnts[4] = {tp, tn, fp, vs};
    unsigned tot[4];
    #pragma unroll
    for (int k = 0; k < 4; ++k) {
        su[threadIdx.x] = cnts[k]; __syncthreads();
        for (int off = 128; off > 0; off >>= 1) {
            if (threadIdx.x < off) su[threadIdx.x] += su[threadIdx.x + off];
            __syncthreads();
        }
        tot[k] = su[0]; __syncthreads();
    }

    if (threadIdx.x == 0) {
        int ms = 0;
        #pragma unroll
        for (int b = 0; b < Bx; ++b) ms += seqlen[b];
        const float maskSum = 32.f * (float)ms;   // 2L * sum(seqlen)
        out_scalars[0] = (float)tot[0];                      // tp
        out_scalars[1] = (float)tot[1];                      // tn
        out_scalars[2] = (float)tot[2];                      // fp
        out_scalars[3] = sel / maskSum;                      // startend_loss
        out_scalars[4] = loss / ((float)tot[3] + 1e-6f);     // span_loss
    }
}

// ---------------------------------------------------------------------------
extern "C" void kernel_launch(void* const* d_in, const int* in_sizes, int n_in,
                              void* d_out, int out_size, void* d_ws, size_t ws_size,
                              hipStream_t stream)
{
    const float* x      = (const float*)d_in[0];
    const float* start  = (const float*)d_in[1];
    const float* endp   = (const float*)d_in[2];
    const int*   span   = (const int*)d_in[3];
    const int*   val    = (const int*)d_in[4];
    const int*   seqlen = (const int*)d_in[5];
    const float* W_se   = (const float*)d_in[6];
    const float* b_se   = (const float*)d_in[7];
    const float* W_span = (const float*)d_in[8];
    const float* b_span = (const float*)d_in[9];
    float* out = (float*)d_out;

    // workspace layout (floats): all regions fully written before read
    float*    ws      = (float*)d_ws;
    float*    ws_se   = ws;                               // 131072  (4096x32 logits)
    float*    aT      = ws + 131072;                      // 65536   ([B][L][N], +b_span)
    float*    bbT     = ws + 196608;                      // 65536   ([B][L][N])
    float*    lossP   = ws + 262144;                      // 4096
    unsigned* cntP    = (unsigned*)(ws + 266240);         // 16384 u32
    float*    seLossP = ws + 282624;                      // 128
    (void)in_sizes; (void)n_in; (void)out_size; (void)ws_size;

    gemm_wmma_kernel<<<256, 128, 0, stream>>>(x, W_se, b_se, W_span, b_span,
                                              ws_se, aT, bbT);
    se_loss_kernel<<<128, 256, 0, stream>>>(ws_se, start, endp, seqlen, seLossP);
    span_kernel<<<SPAN_BLOCKS, 256, 0, stream>>>(span, val, aT, bbT, out,
                                                 lossP, cntP);
    final_kernel<<<1, 256, 0, stream>>>(lossP, cntP, seLossP, seqlen,
                                        out + SPAN_ELEMS);
}